// MSELoss_24945170055384
// MI455X (gfx1250) — compile-verified
//
#include <hip/hip_runtime.h>
#include <hip/hip_bf16.h>

// Problem constants (from reference): [bs=16, seq=4096, nsrc=4, dim=257] f32
constexpr int BS   = 16;
constexpr int SEQ  = 4096;
constexpr int NSRC = 4;
constexpr int DIM  = 257;
constexpr int ROW  = NSRC * DIM;        // 1028 floats per (b, s) row, = 257 float4s
constexpr int BC   = 128;               // chunks (blocks) per batch element
constexpr int RPB  = SEQ / BC;          // 32 rows per block
constexpr int NBLOCKS = BS * BC;        // 2048 blocks, pass 1
constexpr int THREADS = 288;            // 9 wave32 waves; threads 0..256 active (257 float4 cols)

typedef __attribute__((ext_vector_type(2))) float v2f;
typedef __attribute__((ext_vector_type(4))) float v4f;   // native clang vector: OK for NT loads
typedef __attribute__((ext_vector_type(8))) float v8f;

// ---------------------------------------------------------------------------
// Pass 1: stream 539 MB, per-block partial sums of squared diff per source.
// Deterministic fixed-tree LDS reduction; partials[beta*4 + src] -> workspace.
// ---------------------------------------------------------------------------
__global__ __launch_bounds__(THREADS) void euclid_partial_kernel(
    const float* __restrict__ pred, const float* __restrict__ gt,
    float* __restrict__ partials) {
  __shared__ float red[NSRC][THREADS];

  const int t    = threadIdx.x;
  const int beta = blockIdx.x;
  const int b     = beta / BC;
  const int chunk = beta % BC;

  const bool active = (t < 257);
  const int c  = active ? t : 0;
  const int e0 = 4 * c;                       // first element offset within row
  int srcA = e0 / DIM;                        // source of first element (0..3)
  const int bnd = (srcA + 1) * DIM;           // boundary to next source
  const float w0 = (e0 + 0 >= bnd) ? 1.f : 0.f;
  const float w1 = (e0 + 1 >= bnd) ? 1.f : 0.f;
  const float w2 = (e0 + 2 >= bnd) ? 1.f : 0.f;
  const float w3 = (e0 + 3 >= bnd) ? 1.f : 0.f;
  const int srcB = (srcA < 3) ? srcA + 1 : 3;

  float accA = 0.f, accB = 0.f;

  size_t rowBase = (size_t)(b * SEQ + chunk * RPB) * (size_t)ROW + (size_t)e0;

  for (int r = 0; r < RPB; ++r) {
    if (active) {
      const v4f* pp = (const v4f*)(pred + rowBase);
      const v4f* qq = (const v4f*)(gt + rowBase);
      if (r + 1 < RPB) {                      // streaming prefetch (global_prefetch_b8)
        __builtin_prefetch(pred + rowBase + ROW, 0, 0);
        __builtin_prefetch(gt + rowBase + ROW, 0, 0);
      }
      v4f p = __builtin_nontemporal_load(pp);   // single-use stream: NT hint, b128
      v4f q = __builtin_nontemporal_load(qq);
      float dx = p.x - q.x, dy = p.y - q.y, dz = p.z - q.z, dw = p.w - q.w;
      float vx = dx * dx, vy = dy * dy, vz = dz * dz, vw = dw * dw;
      float sB = fmaf(vx, w0, fmaf(vy, w1, fmaf(vz, w2, vw * w3)));
      float sAll = (vx + vy) + (vz + vw);
      accB += sB;
      accA += (sAll - sB);
    }
    rowBase += ROW;
  }

  // Scatter per-thread partials into per-source columns (each thread owns col t).
  red[0][t] = 0.f; red[1][t] = 0.f; red[2][t] = 0.f; red[3][t] = 0.f;
  red[srcA][t] += accA;
  red[srcB][t] += accB;
  __syncthreads();

  // Fixed binary tree: 288 -> 144 -> 72 -> 36 -> 18 -> 9, then serial 9.
  int n = THREADS;
  while (n > 9) {
    int h = n >> 1;
    if (t < h) {
      red[0][t] += red[0][t + h];
      red[1][t] += red[1][t + h];
      red[2][t] += red[2][t + h];
      red[3][t] += red[3][t + h];
    }
    __syncthreads();
    n = h;
  }
  if (t == 0) {
    float s0 = 0.f, s1 = 0.f, s2 = 0.f, s3 = 0.f;
    for (int i = 0; i < 9; ++i) {
      s0 += red[0][i]; s1 += red[1][i]; s2 += red[2][i]; s3 += red[3][i];
    }
    v4f outv; outv.x = s0; outv.y = s1; outv.z = s2; outv.w = s3;
    *(v4f*)(partials + 4 * beta) = outv;
  }
}

// ---------------------------------------------------------------------------
// Pass 2: one wave32. Sum 128 chunk-partials per (b,src) group on the matrix
// unit: D += ones(16x4) x P(4x16) via V_WMMA_F32_16X16X4_F32, then sqrt per
// group and a deterministic xor-butterfly lane reduction.
//
// B-matrix (4x16 f32) VGPR layout per ISA 7.12.2: lane l holds column N=l&15;
// lanes 0-15 carry K={0,1} in {v0,v1}, lanes 16-31 carry K={2,3}.
// partials layout: partials[(bn*BC + chunk)*4 + srcn], group g = bn*4+srcn.
// ---------------------------------------------------------------------------
__global__ __launch_bounds__(32) void euclid_final_kernel(
    const float* __restrict__ partials, float* __restrict__ out) {
  const int lane = threadIdx.x;
  const int N = lane & 15;
  const int kBase = (lane < 16) ? 0 : 2;

  float total = 0.f;
  v2f a; a.x = 1.f; a.y = 1.f;                 // A = ones(16x4), identical in all lanes

  for (int g0 = 0; g0 < BS * NSRC; g0 += 16) { // 4 group-tiles of 16
    const int gg = g0 + N;
    const int bn = gg >> 2;
    const int sn = gg & 3;
    const int base = bn * BC * 4 + sn + kBase * 4;

    v8f acc = {};
    for (int j = 0; j < BC / 4; ++j) {         // 32 WMMAs fold 128 chunks
      v2f bv;
      bv.x = partials[base + 16 * j];          // k = kBase
      bv.y = partials[base + 16 * j + 4];      // k = kBase + 1
      acc = __builtin_amdgcn_wmma_f32_16x16x4_f32(
          /*neg_a=*/false, a, /*neg_b=*/false, bv,
          /*c_mod=*/(short)0, acc, /*reuse_a=*/false, /*reuse_b=*/false);
    }
    // Every row of D equals the column sums; take row M=0 (VGPR0).
    float d = sqrtf(acc[0]);
    float contrib = (lane < 16) ? d : 0.f;     // lanes 16..31 are duplicates
    for (int off = 16; off >= 1; off >>= 1)
      contrib += __shfl_xor(contrib, off, 32);
    total += contrib;                          // all lanes now hold tile sum
  }

  if (lane == 0) out[0] = total;
}

extern "C" void kernel_launch(void* const* d_in, const int* in_sizes, int n_in,
                              void* d_out, int out_size, void* d_ws, size_t ws_size,
                              hipStream_t stream) {
  const float* pred = (const float*)d_in[0];
  const float* gt   = (const float*)d_in[1];
  float* out        = (float*)d_out;
  float* partials   = (float*)d_ws;            // NBLOCKS*4 floats = 32 KB

  euclid_partial_kernel<<<NBLOCKS, THREADS, 0, stream>>>(pred, gt, partials);
  euclid_final_kernel<<<1, 32, 0, stream>>>(partials, out);
}